// S4Cell_77678778515820
// MI455X (gfx1250) — compile-verified
//
#include <hip/hip_runtime.h>

// S4 recurrence (N=256, B=4096, L=1024, complex64) reformulated as GEMMs:
//   Y  = X0 @ Wsh^T + U @ T2 (+ d*U folded into T2 diagonal)   [B,L] complex
//   Xf = X0 @ (A^L)^T + U @ VrevT                              [B,N] complex
// where Wsh rows = c*A^(t+1), T2[t,j] = g[t-j] (g_m = c A^m b), VrevT[n,j] = (A^(L-1-j) b)[n].
// Krylov rows and A^L are built with 10 doubling steps (small GEMMs).
// All matmuls use V_WMMA_F32_16X16X4_F32 (full fp32 matrix path on gfx1250).

typedef float v2f __attribute__((ext_vector_type(2)));
typedef float v8f __attribute__((ext_vector_type(8)));

#define NN 256
#define LL 1024
#define BB 4096

// ---------------------------------------------------------------------------
// Generic complex NT-GEMM:  C[m,n] (+)= sum_k A[m,k] * B[n,k]   (complex mul)
// A: planar (Are, Aim); Aim==nullptr -> A is real.
// B: planar (Bre, Bim).
// C: if interleave: Cre = base of complex64 (float2) array, ldc in complex elems.
//    else: planar Cre/Cim, ldc in floats.
// Wave computes a 32x32 tile (2x2 WMMA tiles); block = 4 waves = 64x64 tile.
// Requires: Nn % 64 == 0, K % 4 == 0. M arbitrary (row-clamped loads, guarded stores).
// ---------------------------------------------------------------------------
__global__ __launch_bounds__(128)
void cgemm_nt_wmma(const float* __restrict__ Are, const float* __restrict__ Aim,
                   const float* __restrict__ Bre, const float* __restrict__ Bim,
                   float* __restrict__ Cre, float* __restrict__ Cim,
                   int M, int Nn, int K, int lda, int ldb, int ldc,
                   int beta, int interleave)
{
    const int lane = threadIdx.x & 31;
    const int wave = threadIdx.x >> 5;
    const int h = lane >> 4;            // K half (0/1) for A/B frags
    const int q = lane & 15;            // row-in-tile for A, col-in-tile for B/C
    const int m0 = blockIdx.x * 64 + (wave >> 1) * 32;
    const int n0 = blockIdx.y * 64 + (wave & 1) * 32;

    v8f accRR[2][2] = {};   // ArBr
    v8f accII[2][2] = {};   // AiBi
    v8f accIM[2][2] = {};   // ArBi + AiBr

    // A-frag rows (clamped for ragged M), B-frag rows (Nn always multiple of 64 here)
    int ma0 = m0 + q;        int ma1 = m0 + 16 + q;
    int ra0 = (ma0 < M) ? ma0 : (M - 1);
    int ra1 = (ma1 < M) ? ma1 : (M - 1);
    int nb0 = n0 + q;        int nb1 = n0 + 16 + q;

    const float* pAre0 = Are + (size_t)ra0 * lda + 2 * h;
    const float* pAre1 = Are + (size_t)ra1 * lda + 2 * h;
    const float* pBre0 = Bre + (size_t)nb0 * ldb + 2 * h;
    const float* pBre1 = Bre + (size_t)nb1 * ldb + 2 * h;
    const float* pBim0 = Bim + (size_t)nb0 * ldb + 2 * h;
    const float* pBim1 = Bim + (size_t)nb1 * ldb + 2 * h;

    if (Aim) {
        const float* pAim0 = Aim + (size_t)ra0 * lda + 2 * h;
        const float* pAim1 = Aim + (size_t)ra1 * lda + 2 * h;
        for (int k = 0; k < K; k += 4) {
            v2f ar0 = *(const v2f*)(pAre0 + k);
            v2f ar1 = *(const v2f*)(pAre1 + k);
            v2f ai0 = *(const v2f*)(pAim0 + k);
            v2f ai1 = *(const v2f*)(pAim1 + k);
            v2f br0 = *(const v2f*)(pBre0 + k);
            v2f br1 = *(const v2f*)(pBre1 + k);
            v2f bi0 = *(const v2f*)(pBim0 + k);
            v2f bi1 = *(const v2f*)(pBim1 + k);
            v2f ar[2] = { ar0, ar1 }, ai[2] = { ai0, ai1 };
            v2f br[2] = { br0, br1 }, bi[2] = { bi0, bi1 };
#pragma unroll
            for (int i = 0; i < 2; ++i)
#pragma unroll
                for (int j = 0; j < 2; ++j) {
                    accRR[i][j] = __builtin_amdgcn_wmma_f32_16x16x4_f32(
                        false, ar[i], false, br[j], (short)0, accRR[i][j], false, false);
                    accII[i][j] = __builtin_amdgcn_wmma_f32_16x16x4_f32(
                        false, ai[i], false, bi[j], (short)0, accII[i][j], false, false);
                    accIM[i][j] = __builtin_amdgcn_wmma_f32_16x16x4_f32(
                        false, ar[i], false, bi[j], (short)0, accIM[i][j], false, false);
                    accIM[i][j] = __builtin_amdgcn_wmma_f32_16x16x4_f32(
                        false, ai[i], false, br[j], (short)0, accIM[i][j], false, false);
                }
        }
    } else {
        // Real A (e.g. U): re = ArBr, im = ArBi
        for (int k = 0; k < K; k += 4) {
            v2f ar0 = *(const v2f*)(pAre0 + k);
            v2f ar1 = *(const v2f*)(pAre1 + k);
            v2f br0 = *(const v2f*)(pBre0 + k);
            v2f br1 = *(const v2f*)(pBre1 + k);
            v2f bi0 = *(const v2f*)(pBim0 + k);
            v2f bi1 = *(const v2f*)(pBim1 + k);
            v2f ar[2] = { ar0, ar1 };
            v2f br[2] = { br0, br1 }, bi[2] = { bi0, bi1 };
#pragma unroll
            for (int i = 0; i < 2; ++i)
#pragma unroll
                for (int j = 0; j < 2; ++j) {
                    accRR[i][j] = __builtin_amdgcn_wmma_f32_16x16x4_f32(
                        false, ar[i], false, br[j], (short)0, accRR[i][j], false, false);
                    accIM[i][j] = __builtin_amdgcn_wmma_f32_16x16x4_f32(
                        false, ar[i], false, bi[j], (short)0, accIM[i][j], false, false);
                }
        }
    }

    // Epilogue: C layout = vgpr r, lane -> row m0+16i+r+8h, col n0+16j+q
#pragma unroll
    for (int i = 0; i < 2; ++i)
#pragma unroll
        for (int j = 0; j < 2; ++j)
#pragma unroll
            for (int r = 0; r < 8; ++r) {
                int m = m0 + i * 16 + r + 8 * h;
                int n = n0 + j * 16 + q;
                if (m < M) {
                    float re = accRR[i][j][r] - accII[i][j][r];
                    float im = accIM[i][j][r];
                    if (interleave) {
                        float2* p = (float2*)Cre + (size_t)m * ldc + n;
                        if (beta) { float2 c = *p; re += c.x; im += c.y; }
                        *p = make_float2(re, im);
                    } else {
                        size_t off = (size_t)m * ldc + n;
                        if (beta) { re += Cre[off]; im += Cim[off]; }
                        Cre[off] = re; Cim[off] = im;
                    }
                }
            }
}

// ---------------------------------------------------------------------------
// Seed: Apow = A, ApowT = A^T, W row0 = c (c A^0), V row0 = b (A^0 b)
// ---------------------------------------------------------------------------
__global__ void seed_kernel(const float* __restrict__ abre, const float* __restrict__ abim,
                            const float* __restrict__ bbre, const float* __restrict__ bbim,
                            const float* __restrict__ cbre, const float* __restrict__ cbim,
                            float* Apre, float* Apim, float* ApTre, float* ApTim,
                            float* Wre, float* Wim, float* Vre, float* Vim)
{
    int idx = blockIdx.x * blockDim.x + threadIdx.x;
    if (idx < NN * NN) {
        Apre[idx] = abre[idx];
        Apim[idx] = abim[idx];
        int m = idx >> 8, n = idx & 255;
        ApTre[idx] = abre[n * NN + m];
        ApTim[idx] = abim[n * NN + m];
        if (idx < NN) {
            Wre[idx] = cbre[idx]; Wim[idx] = cbim[idx];
            Vre[idx] = bbre[idx]; Vim[idx] = bbim[idx];
        }
    }
}

// W row 1024 = W[1023] @ A  (one extra Krylov row: c A^1024)
__global__ void wlast_kernel(float* Wre, float* Wim,
                             const float* __restrict__ abre, const float* __restrict__ abim)
{
    int n = threadIdx.x;
    const float* wr = Wre + 1023 * NN;
    const float* wi = Wim + 1023 * NN;
    float sre = 0.f, sim = 0.f;
    for (int m = 0; m < NN; ++m) {
        float ar = abre[m * NN + n], ai = abim[m * NN + n];
        float xr = wr[m], xi = wi[m];
        sre += xr * ar - xi * ai;
        sim += xr * ai + xi * ar;
    }
    Wre[1024 * NN + n] = sre;
    Wim[1024 * NN + n] = sim;
}

// g[m] = W[m] . b  (impulse response c A^m b), m = 0..1023
__global__ void g_kernel(const float* __restrict__ Wre, const float* __restrict__ Wim,
                         const float* __restrict__ bre, const float* __restrict__ bim,
                         float* gre, float* gim)
{
    int m = blockIdx.x * blockDim.x + threadIdx.x;
    if (m < LL) {
        const float* wr = Wre + m * NN;
        const float* wi = Wim + m * NN;
        float sre = 0.f, sim = 0.f;
        for (int n = 0; n < NN; ++n) {
            float br = bre[n], bi = bim[n];
            sre += wr[n] * br - wi[n] * bi;
            sim += wr[n] * bi + wi[n] * br;
        }
        gre[m] = sre; gim[m] = sim;
    }
}

// T2[t][j] = g[t-j] for j<=t else 0; T2[t][t].re += d  (folds the d*u term)
__global__ void t2_kernel(const float* __restrict__ gre, const float* __restrict__ gim,
                          const float* __restrict__ dv, float* Tre, float* Tim)
{
    int idx = blockIdx.x * blockDim.x + threadIdx.x;
    if (idx < LL * LL) {
        int t = idx >> 10, j = idx & 1023;
        float re = 0.f, im = 0.f;
        if (j <= t) {
            re = gre[t - j]; im = gim[t - j];
            if (j == t) re += dv[0];
        }
        Tre[idx] = re; Tim[idx] = im;
    }
}

// VrevT[n][j] = V[L-1-j][n]  (transpose + time reversal of A^m b rows)
__global__ void vrevt_kernel(const float* __restrict__ Vre, const float* __restrict__ Vim,
                             float* Tre, float* Tim)
{
    int idx = blockIdx.x * blockDim.x + threadIdx.x;
    if (idx < NN * LL) {
        int n = idx >> 10, j = idx & 1023;
        Tre[n * LL + j] = Vre[(LL - 1 - j) * NN + n];
        Tim[n * LL + j] = Vim[(LL - 1 - j) * NN + n];
    }
}

// ---------------------------------------------------------------------------
extern "C" void kernel_launch(void* const* d_in, const int* in_sizes, int n_in,
                              void* d_out, int out_size, void* d_ws, size_t ws_size,
                              hipStream_t stream)
{
    const float* ab_re = (const float*)d_in[0];
    const float* ab_im = (const float*)d_in[1];
    const float* bb_re = (const float*)d_in[2];
    const float* bb_im = (const float*)d_in[3];
    const float* cb_re = (const float*)d_in[4];
    const float* cb_im = (const float*)d_in[5];
    const float* d_sc  = (const float*)d_in[6];
    const float* x0_re = (const float*)d_in[7];
    const float* x0_im = (const float*)d_in[8];
    const float* u     = (const float*)d_in[9];

    // Workspace layout (floats)
    float* ws = (float*)d_ws;
    size_t off = 0;
    float* W_re   = ws + off; off += (size_t)(LL + 1) * NN;   // Krylov rows c A^m, m=0..1024
    float* W_im   = ws + off; off += (size_t)(LL + 1) * NN;
    float* V_re   = ws + off; off += (size_t)LL * NN;         // A^m b rows, m=0..1023
    float* V_im   = ws + off; off += (size_t)LL * NN;
    float* ApA_re = ws + off; off += NN * NN;                 // A^k ping
    float* ApA_im = ws + off; off += NN * NN;
    float* ApAT_re= ws + off; off += NN * NN;                 // (A^k)^T ping
    float* ApAT_im= ws + off; off += NN * NN;
    float* ApB_re = ws + off; off += NN * NN;                 // pong buffers
    float* ApB_im = ws + off; off += NN * NN;
    float* ApBT_re= ws + off; off += NN * NN;
    float* ApBT_im= ws + off; off += NN * NN;
    float* T2_re  = ws + off; off += (size_t)LL * LL;         // Toeplitz conv matrix
    float* T2_im  = ws + off; off += (size_t)LL * LL;
    float* VrT_re = ws + off; off += (size_t)NN * LL;         // reversed/transposed V
    float* VrT_im = ws + off; off += (size_t)NN * LL;
    float* g_re   = ws + off; off += LL;
    float* g_im   = ws + off; off += LL;
    (void)ws_size; (void)in_sizes; (void)n_in; (void)out_size;

    float* xf_out = (float*)d_out;                 // [B,N] complex64 interleaved
    float* ys_out = (float*)d_out + 2 * (size_t)BB * NN; // [B,L] complex64 interleaved

    // 1) Seed Krylov state and A powers
    seed_kernel<<<(NN * NN + 255) / 256, 256, 0, stream>>>(
        ab_re, ab_im, bb_re, bb_im, cb_re, cb_im,
        ApA_re, ApA_im, ApAT_re, ApAT_im, W_re, W_im, V_re, V_im);

    // 2) Ten doubling steps: rows k..2k-1 of W and V, then square A^k
    float* Ap_re = ApA_re; float* Ap_im = ApA_im;
    float* ApT_re = ApAT_re; float* ApT_im = ApAT_im;
    float* Bp_re = ApB_re; float* Bp_im = ApB_im;
    float* BpT_re = ApBT_re; float* BpT_im = ApBT_im;
    for (int s = 0; s < 10; ++s) {
        int k = 1 << s;
        dim3 gw((k + 63) / 64, NN / 64);
        // W[k+i] = W[i] @ A^k  == NT(Wold, (A^k)^T)
        cgemm_nt_wmma<<<gw, 128, 0, stream>>>(
            W_re, W_im, ApT_re, ApT_im,
            W_re + (size_t)k * NN, W_im + (size_t)k * NN,
            k, NN, NN, NN, NN, NN, 0, 0);
        // V[k+i] = A^k V[i]  == NT(Vold, A^k)
        cgemm_nt_wmma<<<gw, 128, 0, stream>>>(
            V_re, V_im, Ap_re, Ap_im,
            V_re + (size_t)k * NN, V_im + (size_t)k * NN,
            k, NN, NN, NN, NN, NN, 0, 0);
        // A^(2k) = NT(A^k, (A^k)^T); (A^(2k))^T = NT((A^k)^T, A^k)
        dim3 ga(NN / 64, NN / 64);
        cgemm_nt_wmma<<<ga, 128, 0, stream>>>(
            Ap_re, Ap_im, ApT_re, ApT_im, Bp_re, Bp_im,
            NN, NN, NN, NN, NN, NN, 0, 0);
        cgemm_nt_wmma<<<ga, 128, 0, stream>>>(
            ApT_re, ApT_im, Ap_re, Ap_im, BpT_re, BpT_im,
            NN, NN, NN, NN, NN, NN, 0, 0);
        // swap ping-pong
        float* t;
        t = Ap_re;  Ap_re  = Bp_re;  Bp_re  = t;
        t = Ap_im;  Ap_im  = Bp_im;  Bp_im  = t;
        t = ApT_re; ApT_re = BpT_re; BpT_re = t;
        t = ApT_im; ApT_im = BpT_im; BpT_im = t;
    }
    // Ap_* now holds A^1024 = A^L.

    // 3) Extra Krylov row c A^L, impulse response g, Toeplitz T2, reversed V
    wlast_kernel<<<1, NN, 0, stream>>>(W_re, W_im, ab_re, ab_im);
    g_kernel<<<LL / 256, 256, 0, stream>>>(W_re, W_im, bb_re, bb_im, g_re, g_im);
    t2_kernel<<<(LL * LL) / 256, 256, 0, stream>>>(g_re, g_im, d_sc, T2_re, T2_im);
    vrevt_kernel<<<(NN * LL) / 256, 256, 0, stream>>>(V_re, V_im, VrT_re, VrT_im);

    // 4) Output GEMMs (interleaved complex64 into d_out)
    // Y  = X0 @ Wsh^T   (Wsh = rows 1..1024 of W)
    cgemm_nt_wmma<<<dim3(BB / 64, LL / 64), 128, 0, stream>>>(
        x0_re, x0_im, W_re + NN, W_im + NN,
        ys_out, nullptr, BB, LL, NN, NN, NN, LL, 0, 1);
    // Y += U @ T2^T(NT)  (causal convolution + d*u via diagonal)
    cgemm_nt_wmma<<<dim3(BB / 64, LL / 64), 128, 0, stream>>>(
        u, nullptr, T2_re, T2_im,
        ys_out, nullptr, BB, LL, LL, LL, LL, LL, 1, 1);
    // Xf = X0 @ (A^L)^T
    cgemm_nt_wmma<<<dim3(BB / 64, NN / 64), 128, 0, stream>>>(
        x0_re, x0_im, Ap_re, Ap_im,
        xf_out, nullptr, BB, NN, NN, NN, NN, NN, 0, 1);
    // Xf += U @ VrevT
    cgemm_nt_wmma<<<dim3(BB / 64, NN / 64), 128, 0, stream>>>(
        u, nullptr, VrT_re, VrT_im,
        xf_out, nullptr, BB, NN, LL, LL, LL, NN, 1, 1);
}